// HeteroSAGEBaseline_27685359190066
// MI455X (gfx1250) — compile-verified
//
#include <hip/hip_runtime.h>
#include <math.h>

#define HID 128
static const int NC = 100000, ND = 20000, NG = 50000, EE = 150000, EP = 100000, LL = 2;

typedef __attribute__((ext_vector_type(2))) float v2f;
typedef __attribute__((ext_vector_type(8))) float v8f;
typedef __attribute__((ext_vector_type(4))) int   v4i;

#if defined(__HIP_DEVICE_COMPILE__) && \
    __has_builtin(__builtin_amdgcn_global_load_async_to_lds_b128) && \
    __has_builtin(__builtin_amdgcn_s_wait_asynccnt)
#define HAVE_ASYNC_LDS 1
#endif

__device__ __forceinline__ void atomAddF(float* p, float v) {
    __hip_atomic_fetch_add(p, v, __ATOMIC_RELAXED, __HIP_MEMORY_SCOPE_AGENT);
}

__device__ __forceinline__ float gelu_exact(float x) {
    return 0.5f * x * (1.0f + erff(x * 0.70710678118654752f));
}

#ifdef HAVE_ASYNC_LDS
// CDNA5 async copy: 16B global -> LDS without VGPR round-trip (ASYNCcnt-tracked).
__device__ __forceinline__ void async_copy_b128(const float* gsrc, float* ldst) {
    v4i* tg = (v4i*)gsrc;   // reinterpret (drops const), still generic AS
    v4i* tl = (v4i*)ldst;
    __attribute__((address_space(1))) v4i* gp =
        (__attribute__((address_space(1))) v4i*)tg;   // generic -> global
    __attribute__((address_space(3))) v4i* lp =
        (__attribute__((address_space(3))) v4i*)tl;   // generic -> LDS
    __builtin_amdgcn_global_load_async_to_lds_b128(gp, lp, 0, 0);
}
#endif

// Stage a 16x128 fp32 tile into LDS. All launch sites have N % 16 == 0, so the
// async path (no per-row predication) is safe; fallback path keeps guards.
__device__ __forceinline__ void stage_tile(const float* __restrict__ src, float* As,
                                           int row0, int N) {
    const int tid = threadIdx.x;
#ifdef HAVE_ASYNC_LDS
    for (int i = tid; i < 16 * HID / 4; i += 256) {
        int r = i >> 5, c4 = i & 31;
        async_copy_b128(src + (size_t)(row0 + r) * HID + c4 * 4, As + r * HID + c4 * 4);
    }
    __builtin_amdgcn_s_wait_asynccnt(0);
    __syncthreads();
#else
    for (int i = tid; i < 16 * HID / 4; i += 256) {
        int r = i >> 5, c4 = i & 31;
        float4 v = make_float4(0.f, 0.f, 0.f, 0.f);
        if (row0 + r < N) v = ((const float4*)(src + (size_t)(row0 + r) * HID))[c4];
        ((float4*)(As + r * HID))[c4] = v;
    }
    __syncthreads();
#endif
}

// One 16x16 output tile: C += (ascale*A)(16x128 in LDS) * W(128x128 global row-major).
// A-fragment per ISA 32-bit 16x4 layout: lanes0-15 -> K=0,1 ; lanes16-31 -> K=2,3.
__device__ __forceinline__ v8f wmma_tile_128(const float* As, const float* __restrict__ W,
                                             int colofs, float ascale, v8f c) {
    const int lane = threadIdx.x & 31;
    const int mn = lane & 15;     // row of A fragment / col of B fragment
    const int hh = lane >> 4;     // half selector
    __builtin_prefetch(W + (size_t)colofs, 0, 1);   // global_prefetch_b8
#pragma unroll
    for (int k = 0; k < HID; k += 4) {
        v2f a, b;
        a.x = As[mn * HID + k + 2 * hh] * ascale;
        a.y = As[mn * HID + k + 2 * hh + 1] * ascale;
        b.x = W[(size_t)(k + 2 * hh) * HID + colofs + mn];
        b.y = W[(size_t)(k + 2 * hh + 1) * HID + colofs + mn];
        c = __builtin_amdgcn_wmma_f32_16x16x4_f32(false, a, false, b, (short)0, c, false, false);
    }
    return c;
}

// out = GELU(LN(X @ W + bias; g, beta)); X:[N,128], W:[128,128]
__global__ void proj_ln_gelu_kernel(const float* __restrict__ X, const float* __restrict__ W,
                                    const float* __restrict__ bias, const float* __restrict__ g,
                                    const float* __restrict__ bta, float* __restrict__ out, int N) {
    __shared__ float As[16 * HID];
    __shared__ float Cs[16][HID];
    const int row0 = blockIdx.x * 16;
    const int tid = threadIdx.x;

    stage_tile(X, As, row0, N);

    const int wave = tid >> 5, lane = tid & 31;
    const int colofs = wave * 16;
    v8f c = {};
    c = wmma_tile_128(As, W, colofs, 1.0f, c);

    const int n = lane & 15, hh = lane >> 4;
#pragma unroll
    for (int j = 0; j < 8; j++)
        Cs[j + 8 * hh][colofs + n] = c[j] + bias[colofs + n];
    __syncthreads();

    // LN+GELU: 16 threads per row, 8 cols each; reduce within 16-lane groups
    const int r = tid >> 4, sc = tid & 15;
    float vals[8], s = 0.f;
#pragma unroll
    for (int i = 0; i < 8; i++) { vals[i] = Cs[r][sc + 16 * i]; s += vals[i]; }
#pragma unroll
    for (int o = 8; o >= 1; o >>= 1) s += __shfl_xor(s, o, 16);
    float mu = s * (1.0f / HID), vs = 0.f;
#pragma unroll
    for (int i = 0; i < 8; i++) { float d = vals[i] - mu; vals[i] = d; vs += d * d; }
#pragma unroll
    for (int o = 8; o >= 1; o >>= 1) vs += __shfl_xor(vs, o, 16);
    float inv = rsqrtf(vs * (1.0f / HID) + 1e-5f);
    if (row0 + r < N) {
#pragma unroll
        for (int i = 0; i < 8; i++) {
            int cc = sc + 16 * i;
            float x = vals[i] * inv * g[cc] + bta[cc];
            out[(size_t)(row0 + r) * HID + cc] = gelu_exact(x);
        }
    }
}

// out = X @ W  (no bias / LN)
__global__ void gemm128_kernel(const float* __restrict__ X, const float* __restrict__ W,
                               float* __restrict__ out, int N) {
    __shared__ float As[16 * HID];
    const int row0 = blockIdx.x * 16;
    const int tid = threadIdx.x;
    stage_tile(X, As, row0, N);
    const int wave = tid >> 5, lane = tid & 31;
    const int colofs = wave * 16;
    v8f c = {};
    c = wmma_tile_128(As, W, colofs, 1.0f, c);
    const int n = lane & 15, hh = lane >> 4;
#pragma unroll
    for (int j = 0; j < 8; j++) {
        int r = j + 8 * hh;
        if (row0 + r < N) out[(size_t)(row0 + r) * HID + colofs + n] = c[j];
    }
}

// in-degree counts (depends only on edge list; computed once per edge type)
__global__ void count_kernel(const int* __restrict__ ei, int E, float* __restrict__ cnt) {
    int e = blockIdx.x * blockDim.x + threadIdx.x;
    if (e < E) atomAddF(cnt + ei[E + e], 1.0f);
}

// per-edge feature scatter: agg[dst] += x_src[src]   (one wave32 per edge, float4/lane)
__global__ void scatter_add_kernel(const float* __restrict__ xsrc, const int* __restrict__ ei,
                                   int E, float* __restrict__ agg) {
    int t = blockIdx.x * blockDim.x + threadIdx.x;
    int e = t >> 5, lane = t & 31;
    if (e >= E) return;
    int s = ei[e], d = ei[E + e];
    float4 v = ((const float4*)(xsrc + (size_t)s * HID))[lane];
    float* dp = agg + (size_t)d * HID + lane * 4;
    atomAddF(dp + 0, v.x);
    atomAddF(dp + 1, v.y);
    atomAddF(dp + 2, v.z);
    atomAddF(dp + 3, v.w);
}

// out = (agg/max(cnt,1)) @ Wl + bl + xdst @ Wr   (fused dual-GEMM per 16-row tile;
// mean-normalization folded into the A-fragment read as a per-lane scalar)
__global__ void sage_gemm_kernel(const float* __restrict__ agg, const float* __restrict__ cnt,
                                 const float* __restrict__ xdst, const float* __restrict__ Wl,
                                 const float* __restrict__ bl, const float* __restrict__ Wr,
                                 float* __restrict__ out, int N) {
    __shared__ float As[16 * HID];
    const int row0 = blockIdx.x * 16;
    const int wave = threadIdx.x >> 5, lane = threadIdx.x & 31;
    const int colofs = wave * 16;
    const int n = lane & 15, hh = lane >> 4;

    // per-lane 1/max(cnt,1) for the single A row this lane reads (row = lane&15)
    float cc = (row0 + n < N) ? cnt[row0 + n] : 1.0f;
    float invc = 1.0f / fmaxf(cc, 1.0f);

    // phase 1: aggregate term (scaled in-register)
    stage_tile(agg, As, row0, N);
    v8f c = {};
    c = wmma_tile_128(As, Wl, colofs, invc, c);
    __syncthreads();

    // phase 2: destination self-term
    stage_tile(xdst, As, row0, N);
    c = wmma_tile_128(As, Wr, colofs, 1.0f, c);

#pragma unroll
    for (int j = 0; j < 8; j++) {
        int r = j + 8 * hh;
        if (row0 + r < N) out[(size_t)(row0 + r) * HID + colofs + n] = c[j] + bl[colofs + n];
    }
}

// h = GELU(LN(scale*(c1[+c2]) + h))  in place; one wave32 per row
__global__ void resid_ln_gelu_kernel(const float* __restrict__ c1, const float* __restrict__ c2,
                                     float scale, float* __restrict__ h, const float* __restrict__ g,
                                     const float* __restrict__ bta, int N) {
    int t = blockIdx.x * blockDim.x + threadIdx.x;
    int row = t >> 5, lane = t & 31;
    if (row >= N) return;
    size_t base = (size_t)row * HID;
    float4 v = ((const float4*)(c1 + base))[lane];
    if (c2) {
        float4 w = ((const float4*)(c2 + base))[lane];
        v.x += w.x; v.y += w.y; v.z += w.z; v.w += w.w;
    }
    float4 hv = ((const float4*)(h + base))[lane];
    float x[4];
    x[0] = scale * v.x + hv.x; x[1] = scale * v.y + hv.y;
    x[2] = scale * v.z + hv.z; x[3] = scale * v.w + hv.w;
    float s = x[0] + x[1] + x[2] + x[3];
#pragma unroll
    for (int o = 16; o >= 1; o >>= 1) s += __shfl_xor(s, o, 32);
    float mu = s * (1.0f / HID), vs = 0.f;
#pragma unroll
    for (int i = 0; i < 4; i++) { x[i] -= mu; vs += x[i] * x[i]; }
#pragma unroll
    for (int o = 16; o >= 1; o >>= 1) vs += __shfl_xor(vs, o, 32);
    float inv = rsqrtf(vs * (1.0f / HID) + 1e-5f);
    float4 ov;
    int c0 = lane * 4;
    ov.x = gelu_exact(x[0] * inv * g[c0 + 0] + bta[c0 + 0]);
    ov.y = gelu_exact(x[1] * inv * g[c0 + 1] + bta[c0 + 1]);
    ov.z = gelu_exact(x[2] * inv * g[c0 + 2] + bta[c0 + 2]);
    ov.w = gelu_exact(x[3] * inv * g[c0 + 3] + bta[c0 + 3]);
    ((float4*)(h + base))[lane] = ov;
}

// out[e] = dot(hcW[ei0[e]], hd[ei1[e]]); one wave32 per edge
__global__ void decode_kernel(const float* __restrict__ hcW, const float* __restrict__ hd,
                              const int* __restrict__ ei, int E, float* __restrict__ out) {
    int t = blockIdx.x * blockDim.x + threadIdx.x;
    int e = t >> 5, lane = t & 31;
    if (e >= E) return;
    int a = ei[e], b = ei[E + e];
    float4 u = ((const float4*)(hcW + (size_t)a * HID))[lane];
    float4 w = ((const float4*)(hd + (size_t)b * HID))[lane];
    float s = u.x * w.x + u.y * w.y + u.z * w.z + u.w * w.w;
#pragma unroll
    for (int o = 16; o >= 1; o >>= 1) s += __shfl_xor(s, o, 32);
    if (lane == 0) out[e] = s;
}

static inline int gtiles(int n) { return (n + 15) / 16; }

static void run_conv(const float* src, const float* dst, const int* ei, int E,
                     const float* Wl, const float* bl, const float* Wr,
                     float* agg, const float* cnt, float* conv_out, int Ndst,
                     hipStream_t stream) {
    (void)hipMemsetAsync(agg, 0, (size_t)Ndst * HID * sizeof(float), stream);
    int thr = E * 32;
    scatter_add_kernel<<<(thr + 255) / 256, 256, 0, stream>>>(src, ei, E, agg);
    sage_gemm_kernel<<<gtiles(Ndst), 256, 0, stream>>>(agg, cnt, dst, Wl, bl, Wr, conv_out, Ndst);
}

extern "C" void kernel_launch(void* const* d_in, const int* in_sizes, int n_in,
                              void* d_out, int out_size, void* d_ws, size_t ws_size,
                              hipStream_t stream) {
    const float* x_chem   = (const float*)d_in[0];
    const float* x_dis    = (const float*)d_in[1];
    const float* gene_emb = (const float*)d_in[2];
    const float* proj_W   = (const float*)d_in[3];
    const float* proj_b   = (const float*)d_in[4];
    const float* p_ln_g   = (const float*)d_in[5];
    const float* p_ln_b   = (const float*)d_in[6];
    const float* sage_Wl  = (const float*)d_in[7];
    const float* sage_bl  = (const float*)d_in[8];
    const float* sage_Wr  = (const float*)d_in[9];
    const float* ln_g     = (const float*)d_in[10];
    const float* ln_b     = (const float*)d_in[11];
    const float* W_cd     = (const float*)d_in[12];
    const int* ei_c2d     = (const int*)d_in[13];
    const int* ei_d2c     = (const int*)d_in[14];
    const int* ei_c2g     = (const int*)d_in[15];
    const int* ei_g2c     = (const int*)d_in[16];
    const int* pos_edge   = (const int*)d_in[17];
    const int* neg_edge   = (const int*)d_in[18];
    float* out = (float*)d_out;

    float* ws = (float*)d_ws;
    size_t o = 0;
    float* hc    = ws + o; o += (size_t)NC * HID;
    float* hd    = ws + o; o += (size_t)ND * HID;
    float* hg    = ws + o; o += (size_t)NG * HID;
    float* convA = ws + o; o += (size_t)NC * HID;   // d2c result (later reused as hcW)
    float* convB = ws + o; o += (size_t)NC * HID;   // g2c result
    float* convC = ws + o; o += (size_t)ND * HID;   // c2d result
    float* convD = ws + o; o += (size_t)NG * HID;   // c2g result
    float* agg   = ws + o; o += (size_t)NC * HID;   // shared scratch (max dst = NC)
    float* cntA  = ws + o; o += (size_t)NC;         // d2c in-degrees
    float* cntB  = ws + o; o += (size_t)NC;         // g2c in-degrees
    float* cntC  = ws + o; o += (size_t)ND;         // c2d in-degrees
    float* cntD  = ws + o; o += (size_t)NG;         // c2g in-degrees

    // edge-type in-degrees: independent of layer, compute once per call
    (void)hipMemsetAsync(cntA, 0, (size_t)(2 * NC + ND + NG) * sizeof(float), stream);
    count_kernel<<<(EE + 255) / 256, 256, 0, stream>>>(ei_d2c, EE, cntA);
    count_kernel<<<(EE + 255) / 256, 256, 0, stream>>>(ei_g2c, EE, cntB);
    count_kernel<<<(EE + 255) / 256, 256, 0, stream>>>(ei_c2d, EE, cntC);
    count_kernel<<<(EE + 255) / 256, 256, 0, stream>>>(ei_c2g, EE, cntD);

    // init node states
    (void)hipMemcpyAsync(hg, gene_emb, (size_t)NG * HID * sizeof(float),
                         hipMemcpyDeviceToDevice, stream);
    proj_ln_gelu_kernel<<<gtiles(NC), 256, 0, stream>>>(
        x_chem, proj_W, proj_b, p_ln_g, p_ln_b, hc, NC);
    proj_ln_gelu_kernel<<<gtiles(ND), 256, 0, stream>>>(
        x_dis, proj_W + HID * HID, proj_b + HID, p_ln_g + HID, p_ln_b + HID, hd, ND);

    for (int l = 0; l < LL; l++) {
        const size_t wsz = (size_t)HID * HID, bsz = HID;
        const float* Wl0 = sage_Wl + (size_t)(l * 4 + 0) * wsz;
        const float* Wl1 = sage_Wl + (size_t)(l * 4 + 1) * wsz;
        const float* Wl2 = sage_Wl + (size_t)(l * 4 + 2) * wsz;
        const float* Wl3 = sage_Wl + (size_t)(l * 4 + 3) * wsz;
        const float* Wr0 = sage_Wr + (size_t)(l * 4 + 0) * wsz;
        const float* Wr1 = sage_Wr + (size_t)(l * 4 + 1) * wsz;
        const float* Wr2 = sage_Wr + (size_t)(l * 4 + 2) * wsz;
        const float* Wr3 = sage_Wr + (size_t)(l * 4 + 3) * wsz;
        const float* bl0 = sage_bl + (size_t)(l * 4 + 0) * bsz;
        const float* bl1 = sage_bl + (size_t)(l * 4 + 1) * bsz;
        const float* bl2 = sage_bl + (size_t)(l * 4 + 2) * bsz;
        const float* bl3 = sage_bl + (size_t)(l * 4 + 3) * bsz;

        run_conv(hc, hd, ei_c2d, EE, Wl0, bl0, Wr0, agg, cntC, convC, ND, stream); // c2d
        run_conv(hd, hc, ei_d2c, EE, Wl1, bl1, Wr1, agg, cntA, convA, NC, stream); // d2c
        run_conv(hc, hg, ei_c2g, EE, Wl2, bl2, Wr2, agg, cntD, convD, NG, stream); // c2g
        run_conv(hg, hc, ei_g2c, EE, Wl3, bl3, Wr3, agg, cntB, convB, NC, stream); // g2c

        const float* g0 = ln_g + (size_t)(l * 3 + 0) * bsz;
        const float* g1 = ln_g + (size_t)(l * 3 + 1) * bsz;
        const float* g2 = ln_g + (size_t)(l * 3 + 2) * bsz;
        const float* b0 = ln_b + (size_t)(l * 3 + 0) * bsz;
        const float* b1 = ln_b + (size_t)(l * 3 + 1) * bsz;
        const float* b2 = ln_b + (size_t)(l * 3 + 2) * bsz;

        resid_ln_gelu_kernel<<<(NC * 32 + 255) / 256, 256, 0, stream>>>(
            convA, convB, 0.5f, hc, g0, b0, NC);
        resid_ln_gelu_kernel<<<(ND * 32 + 255) / 256, 256, 0, stream>>>(
            convC, nullptr, 1.0f, hd, g1, b1, ND);
        resid_ln_gelu_kernel<<<(NG * 32 + 255) / 256, 256, 0, stream>>>(
            convD, nullptr, 1.0f, hg, g2, b2, NG);
    }

    // decoder: hcW = hc @ W_cd (reuse convA), then per-edge dot products
    gemm128_kernel<<<gtiles(NC), 256, 0, stream>>>(hc, W_cd, convA, NC);
    decode_kernel<<<(EP * 32 + 255) / 256, 256, 0, stream>>>(convA, hd, pos_edge, EP, out);
    decode_kernel<<<(EP * 32 + 255) / 256, 256, 0, stream>>>(convA, hd, neg_edge, EP, out + EP);
}